// HybridModel_15504831938715
// MI455X (gfx1250) — compile-verified
//
#include <hip/hip_runtime.h>

typedef __attribute__((ext_vector_type(2))) float v2f;
typedef __attribute__((ext_vector_type(4))) float f4;
typedef __attribute__((ext_vector_type(8))) float v8f;

#define NUM_GRAPHS 256
#define N_PER      64
#define N_NODES    16384
#define E_PER      2048
#define E_TOT      524288
#define D_IN       128
#define DD         64
#define CC         8

// CDNA5 async global->LDS copy (ASYNCcnt-tracked, bypasses VGPRs).
// lds_off = low 32 bits of a generic pointer to __shared__ (LDS aperture maps addr[31:0]).
__device__ __forceinline__ void async_g2l_b128(unsigned lds_off, const void* gaddr) {
    asm volatile("global_load_async_to_lds_b128 %0, %1, off"
                 :: "v"(lds_off), "v"((unsigned long long)(uintptr_t)gaddr)
                 : "memory");
}
__device__ __forceinline__ void wait_asynccnt0() {
    asm volatile("s_wait_asynccnt 0x0" ::: "memory");
}

// ---------------- Kernel 1: per-graph encoder (WMMA GEMM) + softmax + centroids ----------
// block = 128 threads (4 waves), one block per graph (nodes are graph-blocked, 64 per graph)
__global__ __launch_bounds__(128)
void k_encode(const float* __restrict__ node_feat,
              const float* __restrict__ W_atom,
              const float* __restrict__ W_score,
              float* __restrict__ ws_x,
              float* __restrict__ ws_nm,
              float* __restrict__ out_centroid,
              float* __restrict__ out_b2c)
{
    __shared__ __align__(16) float lA[N_PER * D_IN];   // node_feat tile [row][k]
    __shared__ __align__(16) float lB[D_IN * DD];      // W_atom [k][n]
    __shared__ __align__(16) float lX[N_PER * 68];     // x tile, padded stride 68 (16B aligned rows)
    __shared__ __align__(16) float lS[N_PER * CC];     // scores -> softmax probs
    __shared__ __align__(16) float lW[DD * CC];        // W_score
    __shared__ float lDen[CC];

    const int g   = blockIdx.x;
    const int tid = threadIdx.x;

    // ---- async stage A (32KB), B (32KB), W_score (2KB) into LDS ----
    {
        const unsigned ldsA = (unsigned)(uintptr_t)lA;
        const unsigned ldsB = (unsigned)(uintptr_t)lB;
        const unsigned ldsW = (unsigned)(uintptr_t)lW;
        const char* srcA = (const char*)(node_feat + (size_t)g * N_PER * D_IN);
        const char* srcB = (const char*)W_atom;
        const char* srcW = (const char*)W_score;
        for (int i = tid; i < N_PER * D_IN / 4; i += 128)
            async_g2l_b128(ldsA + (unsigned)i * 16u, srcA + (size_t)i * 16);
        for (int i = tid; i < D_IN * DD / 4; i += 128)
            async_g2l_b128(ldsB + (unsigned)i * 16u, srcB + (size_t)i * 16);
        if (tid < DD * CC / 4)
            async_g2l_b128(ldsW + (unsigned)tid * 16u, srcW + (size_t)tid * 16);
        wait_asynccnt0();
    }
    __syncthreads();

    // ---- x = node_feat @ W_atom via V_WMMA_F32_16X16X4_F32 ----
    // A 16x4 layout: lanes 0-15 hold (K=0,K=1), lanes 16-31 hold (K=2,K=3), M = lane%16
    // B 4x16 layout (mirrors C/D striping): lanes 0-15 N=0..15 rows K=0/1, lanes 16-31 rows K=2/3
    const int lane = tid & 31;
    const int half = lane >> 4;
    const int lr   = lane & 15;
    const int m0   = (tid >> 5) * 16;   // wave -> 16-row strip

    for (int nt = 0; nt < 4; ++nt) {
        const int n0 = nt * 16;
        v8f acc = {};
        for (int kk = 0; kk < D_IN / 4; ++kk) {
            const int k = kk * 4 + half * 2;
            v2f a, b;
            a.x = lA[(m0 + lr) * D_IN + k];
            a.y = lA[(m0 + lr) * D_IN + k + 1];
            b.x = lB[k * DD + n0 + lr];
            b.y = lB[(k + 1) * DD + n0 + lr];
            acc = __builtin_amdgcn_wmma_f32_16x16x4_f32(false, a, false, b,
                                                        (short)0, acc, false, false);
        }
        // C/D layout: VGPR v -> row m0 + v (+8 for lanes 16-31), col = n0 + lane%16
#pragma unroll
        for (int v = 0; v < 8; ++v)
            lX[(m0 + v + 8 * half) * 68 + n0 + lr] = acc[v];
    }
    __syncthreads();

    // store x tile to workspace (read later by the base_x replicator), coalesced b128
    for (int i = tid; i < N_PER * DD / 4; i += 128) {
        int r = i >> 4, c4 = (i & 15) * 4;
        f4 v = *(const f4*)&lX[r * 68 + c4];
        *(f4*)&ws_x[(size_t)(g * N_PER + r) * DD + c4] = v;
    }

    // ---- scores = x @ W_score (64x64 @ 64x8) ----
    {
        int row = tid & 63;
        int cb  = (tid >> 6) * 4;
        float s0 = 0.f, s1 = 0.f, s2 = 0.f, s3 = 0.f;
        for (int k = 0; k < DD; ++k) {
            float xv = lX[row * 68 + k];
            s0 += xv * lW[k * CC + cb + 0];
            s1 += xv * lW[k * CC + cb + 1];
            s2 += xv * lW[k * CC + cb + 2];
            s3 += xv * lW[k * CC + cb + 3];
        }
        lS[row * CC + cb + 0] = s0;
        lS[row * CC + cb + 1] = s1;
        lS[row * CC + cb + 2] = s2;
        lS[row * CC + cb + 3] = s3;
    }
    __syncthreads();

    // ---- softmax over C=8 per node ----
    if (tid < N_PER) {
        float mx = lS[tid * CC];
#pragma unroll
        for (int c = 1; c < CC; ++c) mx = fmaxf(mx, lS[tid * CC + c]);
        float e[CC];
        float sum = 0.f;
#pragma unroll
        for (int c = 0; c < CC; ++c) { e[c] = __expf(lS[tid * CC + c] - mx); sum += e[c]; }
        float inv = 1.f / sum;
#pragma unroll
        for (int c = 0; c < CC; ++c) lS[tid * CC + c] = e[c] * inv;
    }
    __syncthreads();

    // nm -> workspace + b2c_w (identical for both reps)
    if (tid < N_PER) {
        size_t n = (size_t)g * N_PER + tid;
        f4 p0 = *(const f4*)&lS[tid * CC];
        f4 p1 = *(const f4*)&lS[tid * CC + 4];
        *(f4*)&ws_nm[n * CC]     = p0;
        *(f4*)&ws_nm[n * CC + 4] = p1;
        *(f4*)&out_b2c[n * CC]                      = p0;
        *(f4*)&out_b2c[n * CC + 4]                  = p1;
        *(f4*)&out_b2c[((size_t)N_NODES + n) * CC]     = p0;
        *(f4*)&out_b2c[((size_t)N_NODES + n) * CC + 4] = p1;
    }
    if (tid < CC) {
        float dsum = 0.f;
        for (int r = 0; r < N_PER; ++r) dsum += lS[r * CC + tid];
        lDen[tid] = dsum;
    }
    __syncthreads();

    // ---- centroid_x[(r*8+c)*256+g][d] = (nm^T x)/(den+1e-6), same for r=0,1 ----
    for (int o = tid; o < CC * DD; o += 128) {
        int c = o >> 6, d = o & 63;
        float acc = 0.f;
        for (int r = 0; r < N_PER; ++r) acc += lS[r * CC + c] * lX[r * 68 + d];
        float val = acc / (lDen[c] + 1e-6f);
        out_centroid[(size_t)((0 * CC + c) * NUM_GRAPHS + g) * DD + d] = val;
        out_centroid[(size_t)((1 * CC + c) * NUM_GRAPHS + g) * DD + d] = val;
    }
}

// ---------------- Kernel 2: per-graph centroid-pair soft edge counts (c2c_w) -------------
// S[i,j] = M[min,max] with M[i,j] = sum_e nm[src,i]*nm[dst,j] (i<=j); diag net-unchanged
__global__ __launch_bounds__(256)
void k_c2c(const float* __restrict__ ws_nm,
           const int* __restrict__ edge_index,
           float* __restrict__ out_c2c)
{
    __shared__ __align__(16) float lnm[N_PER * CC];
    __shared__ float lM[36];

    const int g   = blockIdx.x;
    const int tid = threadIdx.x;

    if (tid < N_PER * CC / 4) {
        async_g2l_b128((unsigned)(uintptr_t)lnm + (unsigned)tid * 16u,
                       (const char*)(ws_nm + (size_t)g * N_PER * CC) + (size_t)tid * 16);
    }
    if (tid < 36) lM[tid] = 0.f;
    wait_asynccnt0();
    __syncthreads();

    const int* e0 = edge_index + (size_t)g * E_PER;
    const int* e1 = edge_index + (size_t)E_TOT + (size_t)g * E_PER;
    const int base = g * N_PER;

    float acc[36];
#pragma unroll
    for (int q = 0; q < 36; ++q) acc[q] = 0.f;

    for (int e = tid; e < E_PER; e += 256) {
        int s = e0[e] - base;   // graph-local endpoints
        int t = e1[e] - base;
        float a[CC], b[CC];
#pragma unroll
        for (int c = 0; c < CC; ++c) { a[c] = lnm[s * CC + c]; b[c] = lnm[t * CC + c]; }
        int q = 0;
#pragma unroll
        for (int i = 0; i < CC; ++i)
#pragma unroll
            for (int j = i; j < CC; ++j)
                acc[q++] += a[i] * b[j];
    }

    // wave32 butterfly reduce, then one LDS atomic per wave per entry
#pragma unroll
    for (int q = 0; q < 36; ++q) {
        float v = acc[q];
        for (int off = 16; off > 0; off >>= 1) v += __shfl_down(v, off, 32);
        if ((tid & 31) == 0) atomicAdd(&lM[q], v);
    }
    __syncthreads();

    if (tid < CC * CC) {
        int i = tid >> 3, j = tid & 7;
        int ii = i < j ? i : j;
        int jj = i < j ? j : i;
        int q = ii * CC - ii * (ii - 1) / 2 + (jj - ii);   // upper-tri-with-diag index
        float v = lM[q];
        out_c2c[(size_t)(0 * NUM_GRAPHS + g) * (CC * CC) + tid] = v;
        out_c2c[(size_t)(1 * NUM_GRAPHS + g) * (CC * CC) + tid] = v;
    }
}

// ---------------- Kernel 3: streaming tiler (power-of-two source) ------------------------
// dst[i] = src[i & mask]; non-temporal b128 stores (outputs are never re-read)
__global__ __launch_bounds__(256)
void k_replicate(const f4* __restrict__ src, f4* __restrict__ dst,
                 unsigned int mask, long total)
{
    long i = (long)blockIdx.x * blockDim.x + threadIdx.x;
    const long stride = (long)gridDim.x * blockDim.x;
    for (; i < total; i += stride) {
        f4 v = src[i & (long)mask];
        __builtin_nontemporal_store(v, &dst[i]);
    }
}

// ---------------- launch --------------------------------------------------------------
extern "C" void kernel_launch(void* const* d_in, const int* in_sizes, int n_in,
                              void* d_out, int out_size, void* d_ws, size_t ws_size,
                              hipStream_t stream)
{
    const float* node_feat  = (const float*)d_in[0];   // [16384,128]
    const float* edge_attr  = (const float*)d_in[1];   // [524288,16]
    const float* W_atom     = (const float*)d_in[2];   // [128,64]
    const float* W_score    = (const float*)d_in[3];   // [64,8]
    const int*   edge_index = (const int*)d_in[4];     // [2,524288]

    float* out = (float*)d_out;
    // output tuple, concatenated flat:
    float* out_base     = out;                         // 33,554,432  (x tiled x32)
    float* out_centroid = out + 33554432L;             //    262,144
    float* out_b2b      = out + 33816576L;             // 16,777,216  (edge_attr tiled x2)
    float* out_b2c      = out + 50593792L;             //    262,144  (nm tiled x2)
    float* out_c2c      = out + 50855936L;             //     32,768

    float* ws_x  = (float*)d_ws;                       // 1,048,576 floats (4 MB)
    float* ws_nm = ws_x + (size_t)N_NODES * DD;        //   131,072 floats

    k_encode<<<NUM_GRAPHS, 128, 0, stream>>>(node_feat, W_atom, W_score,
                                             ws_x, ws_nm, out_centroid, out_b2c);
    k_c2c<<<NUM_GRAPHS, 256, 0, stream>>>(ws_nm, edge_index, out_c2c);

    // base_x: 8,388,608 f4 from a 2^18-f4 source; b2b_attr: 4,194,304 f4 from 2^21-f4 source
    k_replicate<<<4096, 256, 0, stream>>>((const f4*)ws_x, (f4*)out_base,
                                          (1u << 18) - 1u, 8388608L);
    k_replicate<<<4096, 256, 0, stream>>>((const f4*)edge_attr, (f4*)out_b2b,
                                          (1u << 21) - 1u, 4194304L);
}